// EfficientAttention_21363167330926
// MI455X (gfx1250) — compile-verified
//
#include <hip/hip_runtime.h>

// ---------------------------------------------------------------------------
// MI455X (gfx1250) implementation: wave32 + V_WMMA_F32_16X16X4_F32, with
// LDS-staged, double-buffered B tiles shared by all 8 waves of a block.
//
// Roofline: 172 GFLOP vs ~140-400 MB HBM traffic => bandwidth bound at any
// matrix precision, so we keep exact fp32.  The LDS staging is about cache
// bandwidth: sharing B across the block raises block-level arithmetic
// intensity from ~6.4 to ~22 flop/byte and removes the per-WMMA
// s_wait_loadcnt 0 serialization seen in the previous revision.
// ---------------------------------------------------------------------------

typedef float v2f __attribute__((ext_vector_type(2)));
typedef float v8f __attribute__((ext_vector_type(8)));

// Generic C = scale * op(A) * op(B) (+ bias) GEMM.
//   AT   : A stored [K][M] (access A[k*lda+m]) instead of [M][K]
//   B_NT : B stored [N][K] (access B[n*ldb+k])  -> x @ W^T case
//   BIAS : add bias[col]
// Wave tile 16(M) x 64(N), block = 8 waves -> 128 x 64 tile.
// B chunk (KC x 64) staged in LDS, double buffered.
template <bool AT, bool B_NT, bool BIAS>
__global__ __launch_bounds__(256)
void wmma_gemm_f32(const float* __restrict__ A, const float* __restrict__ B,
                   const float* __restrict__ bias, float* __restrict__ C,
                   int K, int lda, int ldb, int ldc,
                   long strideA, long strideB, long strideC, float scale) {
  constexpr int KC = 32;                       // K elements per LDS chunk
  // LDS layouts (padded strides to spread banks, kept even for 8B alignment):
  //   B_NT : lds[n][kk], row stride 34  (64 rows)  -> frag read = ds b64
  //   !B_NT: lds[kk][n], row stride 66  (32 rows)  -> staging store = b64
  __shared__ float lds[2][2176];               // max(64*34, 32*66) floats

  const int z = blockIdx.z;
  A += (size_t)z * (size_t)strideA;
  B += (size_t)z * (size_t)strideB;
  C += (size_t)z * (size_t)strideC;

  const int tid  = threadIdx.x;
  const int lane = tid & 31;
  const int wave = tid >> 5;
  const int m0   = blockIdx.x * 128 + wave * 16;   // wave's M origin
  const int n0   = blockIdx.y * 64;                // block's N origin
  const int t15  = lane & 15;                      // A row / B-C col index
  const int hi   = lane >> 4;                      // K-pair group / C row half

  v8f acc0 = {}, acc1 = {}, acc2 = {}, acc3 = {};

  // ---- cooperative B staging: 2048 floats/chunk = 4 float2 per thread ----
  v2f stg[4];
  auto load_chunk = [&](int kbase) {
#pragma unroll
    for (int i = 0; i < 4; ++i) {
      const int id = tid + 256 * i;
      if constexpr (B_NT) {
        const int n = id >> 4, p = id & 15;        // k-pair p
        stg[i] = *(const v2f*)(B + (size_t)(n0 + n) * ldb + kbase + 2 * p);
      } else {
        const int kk = id >> 5, pn = id & 31;      // n-pair pn
        stg[i] = *(const v2f*)(B + (size_t)(kbase + kk) * ldb + n0 + 2 * pn);
      }
    }
  };
  auto store_chunk = [&](int buf) {
#pragma unroll
    for (int i = 0; i < 4; ++i) {
      const int id = tid + 256 * i;
      if constexpr (B_NT) {
        const int n = id >> 4, p = id & 15;
        *(v2f*)&lds[buf][n * 34 + 2 * p] = stg[i];
      } else {
        const int kk = id >> 5, pn = id & 31;
        *(v2f*)&lds[buf][kk * 66 + 2 * pn] = stg[i];
      }
    }
  };

  // ---- prologue: chunk 0 into buffer 0 ----
  load_chunk(0);
  store_chunk(0);
  __syncthreads();

  const int nch = K / KC;
  for (int c = 0; c < nch; ++c) {
    const int kbase = c * KC;

    // overlap: issue global loads for next chunk before computing this one
    if (c + 1 < nch) load_chunk(kbase + KC);

    // batch-load this wave's A strip for the whole chunk (coalesced b64s,
    // issued up front so they pipeline under the WMMA stream)
    v2f afrag[KC / 4];
#pragma unroll
    for (int j = 0; j < KC / 4; ++j) {
      const int ka = kbase + j * 4 + 2 * hi;       // lane holds {ka, ka+1}
      if constexpr (!AT) {
        afrag[j] = *(const v2f*)(A + (size_t)(m0 + t15) * lda + ka);
      } else {
        afrag[j].x = A[(size_t)(ka)     * lda + (m0 + t15)];
        afrag[j].y = A[(size_t)(ka + 1) * lda + (m0 + t15)];
      }
    }

    const float* lb = lds[c & 1];
#pragma unroll
    for (int j = 0; j < KC / 4; ++j) {
      const int ka = j * 4 + 2 * hi;               // within-chunk K offset
      v2f b0, b1, b2, b3;
      if constexpr (B_NT) {
        b0 = *(const v2f*)&lb[(t15 +  0) * 34 + ka];
        b1 = *(const v2f*)&lb[(t15 + 16) * 34 + ka];
        b2 = *(const v2f*)&lb[(t15 + 32) * 34 + ka];
        b3 = *(const v2f*)&lb[(t15 + 48) * 34 + ka];
      } else {
        const float* p0 = &lb[ka * 66 + t15];
        b0.x = p0[0];   b0.y = p0[66];
        b1.x = p0[16];  b1.y = p0[66 + 16];
        b2.x = p0[32];  b2.y = p0[66 + 32];
        b3.x = p0[48];  b3.y = p0[66 + 48];
      }
      const v2f a = afrag[j];
      acc0 = __builtin_amdgcn_wmma_f32_16x16x4_f32(false, a, false, b0, (short)0, acc0, false, false);
      acc1 = __builtin_amdgcn_wmma_f32_16x16x4_f32(false, a, false, b1, (short)0, acc1, false, false);
      acc2 = __builtin_amdgcn_wmma_f32_16x16x4_f32(false, a, false, b2, (short)0, acc2, false, false);
      acc3 = __builtin_amdgcn_wmma_f32_16x16x4_f32(false, a, false, b3, (short)0, acc3, false, false);
    }

    // publish next chunk into the other buffer (the one last read in c-1)
    if (c + 1 < nch) {
      store_chunk((c + 1) & 1);
      __syncthreads();
    }
  }

  // ---- epilogue: VGPR r -> row m0 + r + 8*hi, col n0 + 16*t + t15 ----
  v8f accs[4] = {acc0, acc1, acc2, acc3};
#pragma unroll
  for (int t = 0; t < 4; ++t) {
    const int col = n0 + 16 * t + t15;
    const float badd = BIAS ? bias[col] : 0.0f;
    float* Cp = C + (size_t)(m0 + 8 * hi) * ldc + col;
#pragma unroll
    for (int r = 0; r < 8; ++r) {
      Cp[(size_t)r * ldc] = accs[t][r] * scale + badd;
    }
  }
}

extern "C" void kernel_launch(void* const* d_in, const int* in_sizes, int n_in,
                              void* d_out, int out_size, void* d_ws, size_t ws_size,
                              hipStream_t stream) {
  (void)in_sizes; (void)n_in; (void)out_size; (void)ws_size;

  constexpr int B = 4, S = 4096, D = 1024;
  constexpr int M1 = B * S;                 // 16384 rows for projections
  const float scale = 1.0f / 32.0f;         // 1/sqrt(1024)

  const float* x  = (const float*)d_in[0];
  const float* Wq = (const float*)d_in[1];
  const float* bq = (const float*)d_in[2];
  const float* Wk = (const float*)d_in[3];
  const float* bk = (const float*)d_in[4];
  const float* Wv = (const float*)d_in[5];
  const float* bv = (const float*)d_in[6];
  float* out = (float*)d_out;

  // Workspace: Q | K | V | KV  (3 * 64MB + 16MB = 208MB)
  const size_t SD = (size_t)S * D;          // 4M elements per batch of Q/K/V
  const size_t DD = (size_t)D * D;          // 1M elements per batch of KV
  float* Q  = (float*)d_ws;
  float* Km = Q  + (size_t)B * SD;
  float* Vm = Km + (size_t)B * SD;
  float* KV = Vm + (size_t)B * SD;

  const dim3 blk(256, 1, 1);

  // ---- Stage 1: Q/K/V = x @ W^T + b   (M=16384, N=1024, K=1024) ----------
  const dim3 g1(M1 / 128, D / 64, 1);
  wmma_gemm_f32<false, true, true><<<g1, blk, 0, stream>>>(
      x, Wq, bq, Q,  D, D, D, D, 0, 0, 0, 1.0f);
  wmma_gemm_f32<false, true, true><<<g1, blk, 0, stream>>>(
      x, Wk, bk, Km, D, D, D, D, 0, 0, 0, 1.0f);
  wmma_gemm_f32<false, true, true><<<g1, blk, 0, stream>>>(
      x, Wv, bv, Vm, D, D, D, D, 0, 0, 0, 1.0f);

  // ---- Stage 2: KV[b] = scale * K[b]^T @ V[b]  (M=N=1024, K=4096) --------
  const dim3 g2(D / 128, D / 64, B);
  wmma_gemm_f32<true, false, false><<<g2, blk, 0, stream>>>(
      Km, Vm, nullptr, KV, S, D, D, D, (long)SD, (long)SD, (long)DD, scale);

  // ---- Stage 3: E[b] = Q[b] @ KV[b]  (M=4096, N=1024, K=1024) ------------
  const dim3 g3(S / 128, D / 64, B);
  wmma_gemm_f32<false, false, false><<<g3, blk, 0, stream>>>(
      Q, KV, nullptr, out, D, D, D, D, (long)SD, (long)DD, (long)SD, 1.0f);
}